// Qwen3Attention_9818295238961
// MI455X (gfx1250) — compile-verified
//
#include <hip/hip_runtime.h>

// ---------------------------------------------------------------------------
// Qwen3 decode attention for MI455X (gfx1250, wave32, WMMA + async-LDS DMA)
// ---------------------------------------------------------------------------
// B=16, HID=1024, H=16, KVH=8, HD=128, PAGES=256, PAGE=16, L=4096
// Memory-bound: streams 537MB of KV cache once (~23us floor at 23.3 TB/s).
//  - V_WMMA_F32_16X16X4_F32 for all matrix work (full f32 precision)
//  - GLOBAL_LOAD_ASYNC_TO_LDS_B128 double-buffered KV tile staging (one
//    KV tile == one contiguous 8KB page), synchronized with s_wait_asynccnt
// ---------------------------------------------------------------------------

typedef __attribute__((ext_vector_type(2))) float v2f;
typedef __attribute__((ext_vector_type(8))) float v8f;

#define NB     16
#define HIDDIM 1024
#define NH     16
#define NKVH   8
#define HDIM   128
#define NPAGES 256
#define PAGESZ 16
#define SEQL   4096   // NPAGES * PAGESZ

static __device__ __forceinline__ v8f wmma_f32(v2f a, v2f b, v8f c) {
  // D(16x16) = A(16x4) * B(4x16) + C   (f32, wave32)
  return __builtin_amdgcn_wmma_f32_16x16x4_f32(
      /*neg_a=*/false, a, /*neg_b=*/false, b,
      /*c_mod=*/(short)0, c, /*reuse_a=*/false, /*reuse_b=*/false);
}

// Issue N async 512B copies (global -> LDS), contiguous, 16B per lane.
// Same INST_OFFSET applies to both global and LDS addresses (ISA 10.x).
template <int I, int N>
struct AsyncCopy {
  static __device__ __forceinline__ void run(unsigned lds, const float* g) {
    asm volatile("global_load_async_to_lds_b128 %0, %1, off offset:%c2"
                 :: "v"(lds), "v"(g), "i"(I * 512) : "memory");
    AsyncCopy<I + 1, N>::run(lds, g);
  }
};
template <int N>
struct AsyncCopy<N, N> {
  static __device__ __forceinline__ void run(unsigned, const float*) {}
};

#define WAIT_ASYNC(n) asm volatile("s_wait_asynccnt " #n ::: "memory")

// ---------------------------------------------------------------------------
// Kernel 1: qkv = x @ [wq | wk | wv]   -> qkv_ws (16 x 4096)
//   cols [0,2048)=q, [2048,3072)=k, [3072,4096)=v
// One wave per 16-col output tile; M = 16 batch rows fits one WMMA tile.
// ---------------------------------------------------------------------------
__global__ __launch_bounds__(256) void qkv_gemm_kernel(
    const float* __restrict__ x,
    const float* __restrict__ wq, const float* __restrict__ wk,
    const float* __restrict__ wv, float* __restrict__ qkv) {
  const int wid  = blockIdx.x * 8 + (threadIdx.x >> 5);
  const int lane = threadIdx.x & 31;
  const int lm   = lane & 15, half = lane >> 4;
  const int n0   = wid * 16;
  const float* bsrc; int ldb, ncol;
  if (n0 < 2048)      { bsrc = wq; ldb = 2048; ncol = n0; }
  else if (n0 < 3072) { bsrc = wk; ldb = 1024; ncol = n0 - 2048; }
  else                { bsrc = wv; ldb = 1024; ncol = n0 - 3072; }
  v8f c = {0.f,0.f,0.f,0.f,0.f,0.f,0.f,0.f};
  const float* arow = x + (size_t)lm * HIDDIM;   // A: M(row)=batch=lm
  #pragma unroll 8
  for (int kk = 0; kk < HIDDIM; kk += 4) {
    const int k0 = kk + 2 * half;                // A layout: half picks K+2
    v2f a = *(const v2f*)(arow + k0);
    v2f b; b.x = bsrc[(size_t)k0       * ldb + ncol + lm];
           b.y = bsrc[(size_t)(k0 + 1) * ldb + ncol + lm];
    c = wmma_f32(a, b, c);
  }
  #pragma unroll
  for (int r = 0; r < 8; ++r)                    // D: N=lm, M=r+8*half
    qkv[(size_t)(r + 8 * half) * 4096 + n0 + lm] = c[r];
}

// ---------------------------------------------------------------------------
// Kernel 2: in-place RMSNorm + RoPE on q rows (256) and k rows (128).
// One wave per 128-elem row; lane holds hd = {l, l+32, l+64, l+96} so RoPE
// pairs (hd, hd+64) stay inside the lane. Folds 1/sqrt(HD) into q.
// ---------------------------------------------------------------------------
__global__ __launch_bounds__(256) void norm_rope_kernel(
    float* __restrict__ qkv,
    const float* __restrict__ qn_w, const float* __restrict__ kn_w,
    const float* __restrict__ cos_t, const float* __restrict__ sin_t,
    const int* __restrict__ start_pos) {
  const int rid  = blockIdx.x * 8 + (threadIdx.x >> 5);
  const int lane = threadIdx.x & 31;
  float* row; const float* w; float oscale; int b;
  if (rid < 256) {                                   // q rows
    b = rid >> 4; const int h = rid & 15;
    row = qkv + (size_t)b * 4096 + (size_t)h * HDIM;
    w = qn_w; oscale = 0.08838834764831845f;         // 1/sqrt(128)
  } else {                                           // k rows
    const int r2 = rid - 256; b = r2 >> 3; const int kvh = r2 & 7;
    row = qkv + (size_t)b * 4096 + 2048 + (size_t)kvh * HDIM;
    w = kn_w; oscale = 1.0f;
  }
  const int pos = start_pos[b];
  float x0 = row[lane], x1 = row[lane + 32], x2 = row[lane + 64], x3 = row[lane + 96];
  float ss = x0*x0 + x1*x1 + x2*x2 + x3*x3;
  #pragma unroll
  for (int off = 16; off >= 1; off >>= 1) ss += __shfl_xor(ss, off, 32);
  const float inv = rsqrtf(ss * (1.0f / 128.0f) + 1e-6f);
  const float n0 = x0 * inv * w[lane],      n1 = x1 * inv * w[lane + 32];
  const float n2 = x2 * inv * w[lane + 64], n3 = x3 * inv * w[lane + 96];
  const float c0 = cos_t[pos * 64 + lane],      s0 = sin_t[pos * 64 + lane];
  const float c1 = cos_t[pos * 64 + lane + 32], s1 = sin_t[pos * 64 + lane + 32];
  row[lane]      = (n0 * c0 - n2 * s0) * oscale;
  row[lane + 32] = (n1 * c1 - n3 * s1) * oscale;
  row[lane + 64] = (n2 * c0 + n0 * s0) * oscale;
  row[lane + 96] = (n3 * c1 + n1 * s1) * oscale;
}

// ---------------------------------------------------------------------------
// Kernel 3: flash-decode attention. Block = (b, kvh), 8 waves.
// Each wave owns tiles t = w, w+8, ... (one tile == one 8KB KV page) and
// double-buffers K/V tiles in LDS via async DMA. Computes
//   S^T = K(16x128) * Q^T(128x16)  (head on lane dim, key on VGPR dim),
// online softmax per lane, then o^T(hd x head) += V^T * P^T via WMMA with the
// P^T B-layout built from 4 shuffles per chunk. Cross-wave merge in LDS.
// Wave 0 additionally processes the current token as a masked broadcast tile.
//
// Dynamic LDS layout (floats):
//   [0,2048)        qs      Q^T: qs[hd*16+head], heads>=2 zero
//   [2048,4096)     o_red   [8][2][128]
//   [4096,4112)     m_red   [8][2]
//   [4112,4128)     l_red   [8][2]
//   [4128,69664)    stage   8 waves * 2 bufs * (2048 K + 2048 V)
// Total 278656 bytes (< 320KB WGP LDS).
// ---------------------------------------------------------------------------
__global__ __launch_bounds__(256) void attn_kernel(
    const float* __restrict__ qkv,
    const float* __restrict__ k_pages, const float* __restrict__ v_pages,
    const int* __restrict__ seq_lens, float* __restrict__ y_ws) {
  extern __shared__ float smem[];
  float* qs    = smem;             // 2048
  float* o_red = smem + 2048;      // 2048
  float* m_red = smem + 4096;      // 16
  float* l_red = smem + 4112;      // 16
  float* stage = smem + 4128;      // 65536

  const int blk = blockIdx.x;
  const int b = blk >> 3, kvh = blk & 7;
  const int tid = threadIdx.x;
  const int w = tid >> 5, lane = tid & 31;
  const int lm = lane & 15, half = lane >> 4;

  const float* qrow = qkv + (size_t)b * 4096 + (size_t)(2 * kvh) * HDIM;
  for (int i = tid; i < HDIM * 16; i += 256) {
    const int col = i & 15, hd = i >> 4;
    qs[i] = (col < 2) ? qrow[col * HDIM + hd] : 0.0f;
  }
  const int sl = seq_lens[b];
  int lens = (sl - 1) & (PAGESZ - 1);
  if (lens == 0) lens = PAGESZ;
  const int valid = (NPAGES - 1) * PAGESZ + lens;  // in (4080, 4096]
  __syncthreads();

  const float* kcur = qkv + (size_t)b * 4096 + 2048 + (size_t)kvh * HDIM;
  const float* vcur = qkv + (size_t)b * 4096 + 3072 + (size_t)kvh * HDIM;
  const size_t PSTR  = (size_t)NB * NKVH * PAGESZ * HDIM;          // page stride
  const size_t bkoff = (size_t)(b * NKVH + kvh) * (PAGESZ * HDIM); // (b,kvh) offset
  const float* kpage = k_pages + bkoff + lane * 4;   // + page*PSTR per tile
  const float* vpage = v_pages + bkoff + lane * 4;

  // Per-wave double buffers: buf c -> K at kbuf[c], V at kbuf[c]+2048 floats.
  float* kbuf[2];
  kbuf[0] = stage + (size_t)(w * 2 + 0) * 4096;
  kbuf[1] = stage + (size_t)(w * 2 + 1) * 4096;
  unsigned klds[2], vlds[2];                          // per-lane LDS byte addrs
  klds[0] = (unsigned)(uintptr_t)(kbuf[0]) + lane * 16;
  klds[1] = (unsigned)(uintptr_t)(kbuf[1]) + lane * 16;
  vlds[0] = klds[0] + 2048 * 4;
  vlds[1] = klds[1] + 2048 * 4;

  float m = -1e30f, lsum = 0.0f;
  const v8f vzero = {0.f,0.f,0.f,0.f,0.f,0.f,0.f,0.f};
  v8f o[8];
  #pragma unroll
  for (int nt = 0; nt < 8; ++nt) o[nt] = vzero;

  // Prologue: stage first tile (t = w < 256, always a normal page).
  AsyncCopy<0, 16>::run(klds[0], kpage + (size_t)w * PSTR);
  AsyncCopy<0, 16>::run(vlds[0], vpage + (size_t)w * PSTR);

  int cb = 0;
  // tiles 0..255 = cached pages (tile tau -> wave tau%8); tile 256 = current token (wave 0)
  for (int t = w; t <= 256; t += 8) {
    const bool cur = (t == 256);
    const int nt_ = t + 8;
    // Stage next tile into the other buffer, then wait for current buffer.
    if (nt_ < 256) {
      AsyncCopy<0, 16>::run(klds[1 - cb], kpage + (size_t)nt_ * PSTR);
      AsyncCopy<0, 16>::run(vlds[1 - cb], vpage + (size_t)nt_ * PSTR);
      WAIT_ASYNC(32);
    } else if (nt_ == 256) {       // only wave 0 reaches this
      AsyncCopy<0, 1>::run(klds[1 - cb], kcur);   // k_cur -> row 0
      AsyncCopy<0, 1>::run(vlds[1 - cb], vcur);   // v_cur -> row 0
      WAIT_ASYNC(2);
    } else {
      WAIT_ASYNC(0);
    }
    const float* kT = kbuf[cb];
    const float* vT = kbuf[cb] + 2048;

    // ---- scores: S^T(key x head) ----
    v8f c = vzero;
    const float* krow = kT + lm * HDIM;          // A: M(row)=key=lm
    #pragma unroll
    for (int kk = 0; kk < HDIM; kk += 4) {
      const int hd0 = kk + 2 * half;             // A: K-dim = hd
      v2f a = *(const v2f*)(krow + hd0);         // ds_load_b64
      v2f bq; bq.x = qs[hd0 * 16 + lm];          // B: Q^T[hd][head=lm]
              bq.y = qs[(hd0 + 1) * 16 + lm];
      c = wmma_f32(a, bq, c);
    }
    // ---- mask ----
    if (cur) {
      #pragma unroll
      for (int r = 0; r < 8; ++r)
        if ((r + 8 * half) != 0) c[r] = -1e30f;  // only key row 0 is real
    } else if (t * 16 + 16 > valid) {            // only last cached tile partial
      #pragma unroll
      for (int r = 0; r < 8; ++r)
        if (t * 16 + r + 8 * half >= valid) c[r] = -1e30f;
    }
    // ---- online softmax (state per head = per lane column) ----
    float tmax = c[0];
    #pragma unroll
    for (int r = 1; r < 8; ++r) tmax = fmaxf(tmax, c[r]);
    tmax = fmaxf(tmax, __shfl_xor(tmax, 16, 32));
    const float newm = fmaxf(m, tmax);
    const float scale = __expf(m - newm);
    float p[8], psum = 0.0f;
    #pragma unroll
    for (int r = 0; r < 8; ++r) { p[r] = __expf(c[r] - newm); psum += p[r]; }
    psum += __shfl_xor(psum, 16, 32);
    lsum = lsum * scale + psum;
    m = newm;
    #pragma unroll
    for (int nt = 0; nt < 8; ++nt) {
      #pragma unroll
      for (int r = 0; r < 8; ++r) o[nt][r] *= scale;
    }
    // ---- PV: o^T(hd x head) += V^T(hd x key) * P^T(key x head) ----
    #pragma unroll
    for (int kc = 0; kc < 4; ++kc) {
      const int hi   = kc >> 1;                  // source half of key rows
      const int base = (kc & 1) * 4;             // source VGPR of key rows
      const int src  = lm + 16 * hi;
      const float t0 = __shfl(p[base + 0], src, 32);
      const float t1 = __shfl(p[base + 2], src, 32);
      const float u0 = __shfl(p[base + 1], src, 32);
      const float u1 = __shfl(p[base + 3], src, 32);
      v2f bb; bb.x = half ? t1 : t0; bb.y = half ? u1 : u0;  // B-layout P^T chunk
      const int vrow = cur ? 0 : (kc * 4 + 2 * half);        // stale rows have p==0
      const float* vr0 = vT + vrow * HDIM;
      const float* vr1 = cur ? vr0 : (vr0 + HDIM);
      #pragma unroll
      for (int nt = 0; nt < 8; ++nt) {
        v2f a; a.x = vr0[nt * 16 + lm];          // ds_read2_b32 (stride 128 dw)
               a.y = vr1[nt * 16 + lm];
        o[nt] = wmma_f32(a, bb, o[nt]);
      }
    }
    cb ^= 1;
  }

  // ---- cross-wave merge via LDS ----
  if (lm < 2) {                                  // lanes 0,1,16,17 hold real heads
    if (half == 0) { m_red[w * 2 + lm] = m; l_red[w * 2 + lm] = lsum; }
    #pragma unroll
    for (int nt = 0; nt < 8; ++nt) {
      #pragma unroll
      for (int r = 0; r < 8; ++r)
        o_red[(w * 2 + lm) * HDIM + nt * 16 + r + 8 * half] = o[nt][r];
    }
  }
  __syncthreads();
  {
    const int head = tid >> 7, hd = tid & 127;   // 256 threads = 2 heads x 128 hd
    float M = m_red[head];
    #pragma unroll
    for (int wv = 1; wv < 8; ++wv) M = fmaxf(M, m_red[wv * 2 + head]);
    float den = 0.0f, acc = 0.0f;
    #pragma unroll
    for (int wv = 0; wv < 8; ++wv) {
      const float e = __expf(m_red[wv * 2 + head] - M);
      den += e * l_red[wv * 2 + head];
      acc += e * o_red[(wv * 2 + head) * HDIM + hd];
    }
    y_ws[(size_t)b * 2048 + (size_t)(2 * kvh + head) * HDIM + hd] = acc / den;
  }
}

// ---------------------------------------------------------------------------
// Kernel 4: out = y_ws(16x2048) @ wo(2048x1024)
// ---------------------------------------------------------------------------
__global__ __launch_bounds__(256) void out_gemm_kernel(
    const float* __restrict__ y, const float* __restrict__ wo,
    float* __restrict__ out) {
  const int wid  = blockIdx.x * 8 + (threadIdx.x >> 5);
  const int lane = threadIdx.x & 31;
  const int lm   = lane & 15, half = lane >> 4;
  const int n0   = wid * 16;
  v8f c = {0.f,0.f,0.f,0.f,0.f,0.f,0.f,0.f};
  const float* arow = y + (size_t)lm * 2048;
  #pragma unroll 8
  for (int kk = 0; kk < 2048; kk += 4) {
    const int k0 = kk + 2 * half;
    v2f a = *(const v2f*)(arow + k0);
    v2f b; b.x = wo[(size_t)k0       * 1024 + n0 + lm];
           b.y = wo[(size_t)(k0 + 1) * 1024 + n0 + lm];
    c = wmma_f32(a, b, c);
  }
  #pragma unroll
  for (int r = 0; r < 8; ++r)
    out[(size_t)(r + 8 * half) * 1024 + n0 + lm] = c[r];
}

// ---------------------------------------------------------------------------
// Host launcher
// Inputs (setup_inputs order): x, k_pages, v_pages, wq, wk, wv, wo,
//                              qn_w, kn_w, cos, sin, seq_lens, start_positions
// ---------------------------------------------------------------------------
extern "C" void kernel_launch(void* const* d_in, const int* in_sizes, int n_in,
                              void* d_out, int out_size, void* d_ws, size_t ws_size,
                              hipStream_t stream) {
  (void)in_sizes; (void)n_in; (void)out_size; (void)ws_size;
  const float* x       = (const float*)d_in[0];
  const float* k_pages = (const float*)d_in[1];
  const float* v_pages = (const float*)d_in[2];
  const float* wq      = (const float*)d_in[3];
  const float* wk      = (const float*)d_in[4];
  const float* wv      = (const float*)d_in[5];
  const float* wo      = (const float*)d_in[6];
  const float* qn_w    = (const float*)d_in[7];
  const float* kn_w    = (const float*)d_in[8];
  const float* cos_t   = (const float*)d_in[9];
  const float* sin_t   = (const float*)d_in[10];
  const int*   seq_lens  = (const int*)d_in[11];
  const int*   start_pos = (const int*)d_in[12];
  float* out = (float*)d_out;

  float* qkv_ws = (float*)d_ws;               // 16 * 4096 floats (256 KB)
  float* y_ws   = qkv_ws + (size_t)NB * 4096; // 16 * 2048 floats (128 KB)

  const size_t attn_lds = (size_t)(4128 + 8 * 2 * 4096) * sizeof(float); // 278656 B

  qkv_gemm_kernel<<<32, 256, 0, stream>>>(x, wq, wk, wv, qkv_ws);
  norm_rope_kernel<<<48, 256, 0, stream>>>(qkv_ws, qn_w, kn_w, cos_t, sin_t, start_pos);
  attn_kernel<<<NB * NKVH, 256, attn_lds, stream>>>(qkv_ws, k_pages, v_pages, seq_lens, y_ws);
  out_gemm_kernel<<<8, 256, 0, stream>>>(y_ws, wo, out);
}